// AttentionModule_2456721293570
// MI455X (gfx1250) — compile-verified
//
#include <hip/hip_runtime.h>
#include <math.h>

// ---------------------------------------------------------------------------
// scatter_mean -> tanh(mean @ W) -> per-node sigmoid attention -> scatter_add
// MI455X (gfx1250): bandwidth-bound (x = 512MB, read twice). One workgroup per
// sorted segment fuses both passes so the second read hits L2 (~125KB/segment
// << 192MB). Main loops use b128 (float4) loads: 512B per wave per issue.
// mean @ W runs on the matrix unit as a padded 16x64 @ 64x64 f32 WMMA tile
// (V_WMMA_F32_16X16X4_F32) -- exact f32 precision, no exec-masked loads.
// ---------------------------------------------------------------------------

typedef float v2f __attribute__((ext_vector_type(2)));
typedef float v8f __attribute__((ext_vector_type(8)));

constexpr int DIM   = 64;   // feature dim
constexpr int NWAVE = 8;    // waves per block (wave32)
constexpr int BLOCK = 256;  // threads per block

// Segment boundaries via binary search over the sorted batch array.
__global__ void seg_bounds_kernel(const int* __restrict__ batch, int n, int nseg,
                                  int* __restrict__ offs) {
  int b = blockIdx.x * blockDim.x + threadIdx.x;
  if (b > nseg) return;
  if (b == nseg) { offs[nseg] = n; return; }
  int lo = 0, hi = n;                       // lower_bound(batch, b)
  while (lo < hi) {
    int mid = (lo + hi) >> 1;
    if (batch[mid] < b) lo = mid + 1; else hi = mid;
  }
  offs[b] = lo;
}

__global__ __launch_bounds__(BLOCK)
void fused_attention_kernel(const float* __restrict__ x,
                            const int* __restrict__ offs,
                            const float* __restrict__ Wm,
                            float* __restrict__ out) {
  __shared__ float mean_s[DIM];
  __shared__ float g_s[DIM];
  __shared__ float partial[NWAVE * DIM];

  const int b    = blockIdx.x;
  const int tid  = threadIdx.x;
  const int w    = tid >> 5;        // wave id 0..7
  const int l    = tid & 31;        // lane id (wave32)
  const int hl   = l >> 4;          // half-wave: 0 -> row r, 1 -> row r+1
  const int c    = l & 15;          // column group: cols 4c..4c+3
  const int start = offs[b];
  const int end   = offs[b + 1];
  const int cnt   = end - start;
  const float4* xv = (const float4*)x;   // row = 16 float4s

  // ---- pass 1: column sums (HBM stream, b128 loads, 2 rows/wave/issue) ----
  float4 s4 = make_float4(0.f, 0.f, 0.f, 0.f);
  for (int rb = start + 2 * w; rb < end; rb += 2 * NWAVE) {
    const int r = rb + hl;
    __builtin_prefetch(xv + (size_t)(r + 4 * NWAVE) * 16 + c, 0, 0); // global_prefetch_b8
    float4 v = (r < end) ? xv[(size_t)r * 16 + c]
                         : make_float4(0.f, 0.f, 0.f, 0.f);
    s4.x += v.x; s4.y += v.y; s4.z += v.z; s4.w += v.w;
  }
  // fold the two half-wave row streams (same columns) into lanes 0..15
  s4.x += __shfl_xor(s4.x, 16, 32);
  s4.y += __shfl_xor(s4.y, 16, 32);
  s4.z += __shfl_xor(s4.z, 16, 32);
  s4.w += __shfl_xor(s4.w, 16, 32);
  if (hl == 0) {
    partial[w * DIM + 4 * c + 0] = s4.x;
    partial[w * DIM + 4 * c + 1] = s4.y;
    partial[w * DIM + 4 * c + 2] = s4.z;
    partial[w * DIM + 4 * c + 3] = s4.w;
  }
  __syncthreads();
  if (tid < DIM) {
    float s = 0.f;
#pragma unroll
    for (int i = 0; i < NWAVE; ++i) s += partial[i * DIM + tid];
    mean_s[tid] = s / ((cnt > 0) ? (float)cnt : 1.0f);
  }
  __syncthreads();

  // ---- g = tanh(mean @ W): padded 16x64 @ 64x64 tile on the matrix unit ----
#if __has_builtin(__builtin_amdgcn_wmma_f32_16x16x4_f32)
  if (tid < 32) {
    // A 16x4 f32 layout: lane L -> M = L%16, K = 2*(L/16) + vgpr. Only M=0 valid.
    // B 4x16 f32 layout: lane L -> N = L%16, K = 2*(L/16) + vgpr.
    const float amask = (c == 0) ? 1.0f : 0.0f;  // zero A rows M != 0 w/o branches
    v8f acc0 = {}, acc1 = {}, acc2 = {}, acc3 = {};
#pragma unroll 4
    for (int kk = 0; kk < 16; ++kk) {            // K = 64 in steps of 4
      const int kb = kk * 4 + hl * 2;
      const float mx = mean_s[kb];               // contiguous -> ds_load_b64
      const float my = mean_s[kb + 1];
      v2f a;
      a.x = mx * amask;                          // unconditional select
      a.y = my * amask;
      v2f b0, b1, b2, b3;
      b0.x = Wm[kb * DIM + c];        b0.y = Wm[(kb + 1) * DIM + c];
      b1.x = Wm[kb * DIM + 16 + c];   b1.y = Wm[(kb + 1) * DIM + 16 + c];
      b2.x = Wm[kb * DIM + 32 + c];   b2.y = Wm[(kb + 1) * DIM + 32 + c];
      b3.x = Wm[kb * DIM + 48 + c];   b3.y = Wm[(kb + 1) * DIM + 48 + c];
      acc0 = __builtin_amdgcn_wmma_f32_16x16x4_f32(false, a, false, b0, (short)0, acc0, false, false);
      acc1 = __builtin_amdgcn_wmma_f32_16x16x4_f32(false, a, false, b1, (short)0, acc1, false, false);
      acc2 = __builtin_amdgcn_wmma_f32_16x16x4_f32(false, a, false, b2, (short)0, acc2, false, false);
      acc3 = __builtin_amdgcn_wmma_f32_16x16x4_f32(false, a, false, b3, (short)0, acc3, false, false);
    }
    // D layout: VGPR0, lanes 0..15 hold row M=0, N=lane.
    if (l < 16) {
      g_s[l]      = tanhf(acc0[0]);
      g_s[16 + l] = tanhf(acc1[0]);
      g_s[32 + l] = tanhf(acc2[0]);
      g_s[48 + l] = tanhf(acc3[0]);
    }
  }
#else
  if (tid < DIM) {
    float s = 0.f;
    for (int k = 0; k < DIM; ++k) s += mean_s[k] * Wm[k * DIM + tid];
    g_s[tid] = tanhf(s);
  }
#endif
  __syncthreads();

  // ---- pass 2: coef = sigmoid(x . g), accumulate coef*x (L2-resident) ----
  const float g0 = g_s[4 * c + 0];
  const float g1 = g_s[4 * c + 1];
  const float g2 = g_s[4 * c + 2];
  const float g3 = g_s[4 * c + 3];
  float4 a4 = make_float4(0.f, 0.f, 0.f, 0.f);
  for (int rb = start + 2 * w; rb < end; rb += 2 * NWAVE) {
    const int r = rb + hl;
    float4 v = (r < end) ? xv[(size_t)r * 16 + c]
                         : make_float4(0.f, 0.f, 0.f, 0.f);
    float p = v.x * g0 + v.y * g1 + v.z * g2 + v.w * g3;
#pragma unroll
    for (int m = 8; m >= 1; m >>= 1) p += __shfl_xor(p, m, 32);  // 16-lane half reduce
    float coef = 1.0f / (1.0f + __expf(-p));   // invalid row: v==0 -> coef*0 == 0
    a4.x += coef * v.x;
    a4.y += coef * v.y;
    a4.z += coef * v.z;
    a4.w += coef * v.w;
  }
  a4.x += __shfl_xor(a4.x, 16, 32);
  a4.y += __shfl_xor(a4.y, 16, 32);
  a4.z += __shfl_xor(a4.z, 16, 32);
  a4.w += __shfl_xor(a4.w, 16, 32);
  if (hl == 0) {
    partial[w * DIM + 4 * c + 0] = a4.x;
    partial[w * DIM + 4 * c + 1] = a4.y;
    partial[w * DIM + 4 * c + 2] = a4.z;
    partial[w * DIM + 4 * c + 3] = a4.w;
  }
  __syncthreads();
  if (tid < DIM) {
    float s = 0.f;
#pragma unroll
    for (int i = 0; i < NWAVE; ++i) s += partial[i * DIM + tid];
    out[(size_t)b * DIM + tid] = s;   // every segment writes (zero-count -> 0)
  }
}

extern "C" void kernel_launch(void* const* d_in, const int* in_sizes, int n_in,
                              void* d_out, int out_size, void* d_ws, size_t ws_size,
                              hipStream_t stream) {
  const float* x     = (const float*)d_in[0];
  const int*   batch = (const int*)d_in[1];
  const float* Wm    = (const float*)d_in[2];
  float*       out   = (float*)d_out;

  const int n    = in_sizes[1];          // N nodes
  const int nseg = out_size / DIM;       // B segments

  int* offs = (int*)d_ws;                // [nseg+1] ints in workspace

  const int tb = (nseg + 1 + 255) / 256;
  seg_bounds_kernel<<<tb, 256, 0, stream>>>(batch, n, nseg, offs);
  fused_attention_kernel<<<nseg, BLOCK, 0, stream>>>(x, offs, Wm, out);
}